// LightGCN_88897233092956
// MI455X (gfx1250) — compile-verified
//
#include <hip/hip_runtime.h>
#include <hip/hip_bf16.h>
#include <stdint.h>

#define TPB  256
#define TILE 256

// ---------- CDNA5 async global->LDS copy (ASYNCcnt path) ----------
__device__ __forceinline__ void async_b32(unsigned lds_addr, const void* base,
                                          unsigned byte_off) {
  // GVS mode: saddr = 64-bit SGPR base, vaddr = 32-bit per-lane byte offset,
  // vdst = per-lane LDS byte address.  Tracked with ASYNCcnt.
  asm volatile("global_load_async_to_lds_b32 %0, %1, %2"
               :: "v"(lds_addr), "v"(byte_off), "s"(base)
               : "memory");
}

__device__ __forceinline__ void wait_async0() {
#if defined(__has_builtin) && __has_builtin(__builtin_amdgcn_s_wait_asynccnt)
  __builtin_amdgcn_s_wait_asynccnt(0);
#else
  asm volatile("s_wait_asynccnt 0" ::: "memory");
#endif
}

__device__ __forceinline__ unsigned lds_addr_of(const void* p) {
  // Generic shared pointer = {SHARED_BASE aperture, 32-bit LDS offset};
  // low 32 bits are the LDS byte offset the async instruction wants.
  return (unsigned)(uint64_t)p;
}

// ---------- init: x_cur = acc = concat(user,item); x_next = 0 ----------
__global__ __launch_bounds__(TPB) void lgcn_init(const float4* __restrict__ u,
                                                 const float4* __restrict__ it,
                                                 float4* __restrict__ cur,
                                                 float4* __restrict__ acc,
                                                 float4* __restrict__ nxt,
                                                 int n4u, int n4) {
  int i = blockIdx.x * blockDim.x + threadIdx.x;
  if (i < n4) {
    float4 v = (i < n4u) ? u[i] : it[i - n4u];
    cur[i] = v;
    acc[i] = v;
    nxt[i] = make_float4(0.f, 0.f, 0.f, 0.f);
  }
}

// ---------- SpMM: x_next[r] += v * x_cur[c], edges streamed via async LDS ----------
__global__ __launch_bounds__(TPB) void lgcn_spmm(const int* __restrict__ rows,
                                                 const int* __restrict__ cols,
                                                 const float* __restrict__ vals,
                                                 const float4* __restrict__ xin,
                                                 float* __restrict__ xout,
                                                 int E, int NT) {
  __shared__ int   s_rows[2][TILE];
  __shared__ int   s_cols[2][TILE];
  __shared__ float s_vals[2][TILE];

  const int t      = threadIdx.x;
  const int lane_q = t & 15;        // which float4 of the 64-dim row
  const int e_sub  = t >> 4;        // 16 edges in flight per pass
  const int dim_o  = lane_q << 2;   // dim (float) offset within output row

  int tile = blockIdx.x;
  int buf  = 0;

  if (tile < NT) {                  // prime buffer 0
    int e = tile * TILE + t;
    if (e < E) {
      unsigned off = (unsigned)e << 2;
      async_b32(lds_addr_of(&s_rows[0][t]), rows, off);
      async_b32(lds_addr_of(&s_cols[0][t]), cols, off);
      async_b32(lds_addr_of(&s_vals[0][t]), vals, off);
    }
  }

  for (; tile < NT; tile += gridDim.x) {
    wait_async0();                  // my slice of current buffer landed
    __syncthreads();                // everyone's slice landed

    int ntile = tile + gridDim.x;   // prefetch next tile into other buffer
    if (ntile < NT) {
      int e  = ntile * TILE + t;
      int nb = buf ^ 1;
      if (ntile * TILE + TILE <= E) {          // uniform full-tile path
        unsigned off = (unsigned)e << 2;
        async_b32(lds_addr_of(&s_rows[nb][t]), rows, off);
        async_b32(lds_addr_of(&s_cols[nb][t]), cols, off);
        async_b32(lds_addr_of(&s_vals[nb][t]), vals, off);
      } else if (e < E) {                      // tail tile
        unsigned off = (unsigned)e << 2;
        async_b32(lds_addr_of(&s_rows[nb][t]), rows, off);
        async_b32(lds_addr_of(&s_cols[nb][t]), cols, off);
        async_b32(lds_addr_of(&s_vals[nb][t]), vals, off);
      }
    }

    const int base_e = tile * TILE;
    if (base_e + TILE <= E) {
      // ---- fast path: full tile, no per-lane bound checks ----
#pragma unroll
      for (int p = 0; p < TILE / 16; ++p) {
        int   el = (p << 4) + e_sub;
        int   r  = s_rows[buf][el];
        int   c  = s_cols[buf][el];
        float v  = s_vals[buf][el];
        float4 x = xin[c * 16 + lane_q];       // 256B contiguous per edge
        float* dst = xout + ((int64_t)r << 6) + dim_o;
        (void)__hip_atomic_fetch_add(dst + 0, v * x.x, __ATOMIC_RELAXED, __HIP_MEMORY_SCOPE_AGENT);
        (void)__hip_atomic_fetch_add(dst + 1, v * x.y, __ATOMIC_RELAXED, __HIP_MEMORY_SCOPE_AGENT);
        (void)__hip_atomic_fetch_add(dst + 2, v * x.z, __ATOMIC_RELAXED, __HIP_MEMORY_SCOPE_AGENT);
        (void)__hip_atomic_fetch_add(dst + 3, v * x.w, __ATOMIC_RELAXED, __HIP_MEMORY_SCOPE_AGENT);
      }
    } else {
      // ---- tail tile: guarded ----
      for (int p = 0; p < TILE / 16; ++p) {
        int el = (p << 4) + e_sub;
        if (base_e + el < E) {
          int   r  = s_rows[buf][el];
          int   c  = s_cols[buf][el];
          float v  = s_vals[buf][el];
          float4 x = xin[c * 16 + lane_q];
          float* dst = xout + ((int64_t)r << 6) + dim_o;
          (void)__hip_atomic_fetch_add(dst + 0, v * x.x, __ATOMIC_RELAXED, __HIP_MEMORY_SCOPE_AGENT);
          (void)__hip_atomic_fetch_add(dst + 1, v * x.y, __ATOMIC_RELAXED, __HIP_MEMORY_SCOPE_AGENT);
          (void)__hip_atomic_fetch_add(dst + 2, v * x.z, __ATOMIC_RELAXED, __HIP_MEMORY_SCOPE_AGENT);
          (void)__hip_atomic_fetch_add(dst + 3, v * x.w, __ATOMIC_RELAXED, __HIP_MEMORY_SCOPE_AGENT);
        }
      }
    }
    __syncthreads();                // all reads of buf done before reuse
    buf ^= 1;
  }
}

// ---------- acc = (acc + add) * scale; also zero the next scatter target ----------
__global__ __launch_bounds__(TPB) void lgcn_accum(float4* __restrict__ acc,
                                                  const float4* __restrict__ add,
                                                  float4* __restrict__ zbuf,
                                                  float scale, int n4) {
  int i = blockIdx.x * blockDim.x + threadIdx.x;
  if (i < n4) {
    float4 a = acc[i], b = add[i];
    a.x = (a.x + b.x) * scale;
    a.y = (a.y + b.y) * scale;
    a.z = (a.z + b.z) * scale;
    a.w = (a.w + b.w) * scale;
    acc[i]  = a;
    zbuf[i] = make_float4(0.f, 0.f, 0.f, 0.f);
  }
}

extern "C" void kernel_launch(void* const* d_in, const int* in_sizes, int n_in,
                              void* d_out, int out_size, void* d_ws, size_t ws_size,
                              hipStream_t stream) {
  const float* user = (const float*)d_in[0];
  const float* item = (const float*)d_in[1];
  const int*   rows = (const int*)d_in[2];
  const int*   cols = (const int*)d_in[3];
  const float* vals = (const float*)d_in[4];

  const int U = in_sizes[0] / 64;
  const int I = in_sizes[1] / 64;
  const int N = U + I;
  const int E = in_sizes[2];

  const int n4  = N * 16;   // float4 count of an N x 64 buffer
  const int n4u = U * 16;

  float* w0 = (float*)d_ws;                 // ping
  float* w1 = w0 + (size_t)N * 64;          // pong

  const int eb = (n4 + TPB - 1) / TPB;

  lgcn_init<<<eb, TPB, 0, stream>>>((const float4*)user, (const float4*)item,
                                    (float4*)w0, (float4*)d_out, (float4*)w1,
                                    n4u, n4);

  const int NT   = (E + TILE - 1) / TILE;
  const int grid = NT < 2048 ? NT : 2048;   // cap at 2048 blocks; further tiles are grid-strided

  float* cur = w0;
  float* nxt = w1;
  for (int l = 0; l < 3; ++l) {
    lgcn_spmm<<<grid, TPB, 0, stream>>>(rows, cols, vals,
                                        (const float4*)cur, nxt, E, NT);
    lgcn_accum<<<eb, TPB, 0, stream>>>((float4*)d_out, (const float4*)nxt,
                                       (float4*)cur, (l == 2) ? 0.25f : 1.0f, n4);
    float* tmp = cur; cur = nxt; nxt = tmp;
  }
}